// DGG_StraightThrough_10617159156341
// MI455X (gfx1250) — compile-verified
//
#include <hip/hip_runtime.h>

// Reference reduces to: out[b, n, :] = one_hot(top8(temp[n, :]))  for all b.
// (softmax over a singleton axis makes log_p == 0, so y == temp broadcast;
//  adj == y_hard in the forward pass.)  Memory-bound: ~80MB @ 23.3 TB/s.

#define N_TOK 2048
#define BATCH 4
#define TOPK  8
#define BLOCK 256

#if __has_builtin(__builtin_amdgcn_global_load_async_to_lds_b128) && \
    __has_builtin(__builtin_amdgcn_global_store_async_from_lds_b128)
#define USE_ASYNC_LDS 1
#else
#define USE_ASYNC_LDS 0
#endif

// Builtin signature (from hipcc diagnostic): param0 = v4i in addrspace(1)
// (printed as "__device__"), param1 = v4i in addrspace(3), then imm offset,
// imm cpol:  "vV4i*1V4i*3IiIi".
typedef int v4i __attribute__((ext_vector_type(4)));
typedef __attribute__((address_space(1))) v4i gv4i;
typedef __attribute__((address_space(3))) v4i sv4i;

__device__ __forceinline__ void wait_async_zero() {
#if __has_builtin(__builtin_amdgcn_s_wait_asynccnt)
  __builtin_amdgcn_s_wait_asynccnt(0);
#else
  asm volatile("s_wait_asynccnt 0" ::: "memory");
#endif
}

// Map float bits to an unsigned key that is monotone in float ordering.
__device__ __forceinline__ unsigned int ordkey(float f) {
  unsigned int u = __float_as_uint(f);
  return (u & 0x80000000u) ? ~u : (u | 0x80000000u);
}

__device__ __forceinline__ unsigned long long umax64(unsigned long long a,
                                                     unsigned long long b) {
  return a > b ? a : b;
}

__global__ __launch_bounds__(BLOCK)
void topk_onehot_kernel(const float* __restrict__ temp, float* __restrict__ out) {
  __shared__ __align__(16) float srow[N_TOK];   // staged temp row
  __shared__ __align__(16) float orow[N_TOK];   // one-hot output row
  __shared__ unsigned long long wred[BLOCK / 32];
  __shared__ unsigned long long bkey;
  __shared__ int topidx[TOPK];

  const int row  = blockIdx.x;
  const int t    = threadIdx.x;
  const int lane = t & 31;
  const int wave = t >> 5;

  // ---- Stage temp[row, :] into LDS via the CDNA5 async data mover ----
  const float* gsrc = temp + (size_t)row * N_TOK;
#if USE_ASYNC_LDS
  __builtin_amdgcn_global_load_async_to_lds_b128(
      (gv4i*)(void*)(gsrc + t * 8),     (sv4i*)(void*)(&srow[t * 8]),     0, 0);
  __builtin_amdgcn_global_load_async_to_lds_b128(
      (gv4i*)(void*)(gsrc + t * 8 + 4), (sv4i*)(void*)(&srow[t * 8 + 4]), 0, 0);
  wait_async_zero();
#else
  {
    const float4* g4 = (const float4*)gsrc;
    float4* s4 = (float4*)srow;
    s4[t * 2]     = g4[t * 2];
    s4[t * 2 + 1] = g4[t * 2 + 1];
  }
#endif
  __syncthreads();

  // ---- 8 argmax passes; tie-break toward lower index (matches lax.top_k) ----
  for (int pass = 0; pass < TOPK; ++pass) {
    unsigned long long k = 0ull;  // 0 is below every real key
    for (int j = t; j < N_TOK; j += BLOCK) {
      unsigned long long key =
          ((unsigned long long)ordkey(srow[j]) << 32) | (unsigned int)(~j);
      k = umax64(k, key);
    }
    // wave32 reduction
    for (int off = 16; off; off >>= 1) k = umax64(k, __shfl_xor(k, off, 32));
    if (lane == 0) wred[wave] = k;
    __syncthreads();
    if (wave == 0) {
      unsigned long long kk = (lane < (BLOCK / 32)) ? wred[lane] : 0ull;
      for (int off = 4; off; off >>= 1) kk = umax64(kk, __shfl_xor(kk, off, 32));
      if (lane == 0) bkey = kk;
    }
    __syncthreads();
    if (t == 0) {
      int idx = (int)(~(unsigned int)bkey);   // recover index from low word
      topidx[pass] = idx;
      srow[idx] = __uint_as_float(0xFF800000u);  // -inf: exclude from later passes
    }
    __syncthreads();
  }

  // ---- Build one-hot row in LDS ----
  for (int j = t; j < N_TOK; j += BLOCK) orow[j] = 0.0f;
  __syncthreads();
  if (t < TOPK) orow[topidx[t]] = 1.0f;
  __syncthreads();

  // ---- Stream the row to all 4 batch slices (async LDS->global) ----
  for (int b = 0; b < BATCH; ++b) {
    float* dst = out + ((size_t)b * N_TOK + row) * N_TOK;
#if USE_ASYNC_LDS
    __builtin_amdgcn_global_store_async_from_lds_b128(
        (gv4i*)(void*)(dst + t * 8),     (sv4i*)(void*)(&orow[t * 8]),     0, 0);
    __builtin_amdgcn_global_store_async_from_lds_b128(
        (gv4i*)(void*)(dst + t * 8 + 4), (sv4i*)(void*)(&orow[t * 8 + 4]), 0, 0);
#else
    float4* d4 = (float4*)dst;
    const float4* o4 = (const float4*)orow;
    d4[t * 2]     = o4[t * 2];
    d4[t * 2 + 1] = o4[t * 2 + 1];
#endif
  }
#if USE_ASYNC_LDS
  wait_async_zero();
#endif
}

extern "C" void kernel_launch(void* const* d_in, const int* in_sizes, int n_in,
                              void* d_out, int out_size, void* d_ws, size_t ws_size,
                              hipStream_t stream) {
  (void)in_sizes; (void)n_in; (void)out_size; (void)d_ws; (void)ws_size;
  // setup_inputs order: x[0], temp[1], W_proj[2], b_proj[3], W_dist[4], b_dist[5]
  // Only temp matters: the projection/distance path is dead code in the forward value.
  const float* temp = (const float*)d_in[1];
  float* out = (float*)d_out;
  topk_onehot_kernel<<<dim3(N_TOK), dim3(BLOCK), 0, stream>>>(temp, out);
}